// PersistentHomology_19610820674225
// MI455X (gfx1250) — compile-verified
//
#include <hip/hip_runtime.h>

#define B_      16
#define N_      1024
#define NSTEPS  20
#define NBINS   22   // buckets 0..20 in range, 21 = "beyond max scale"

typedef __attribute__((ext_vector_type(2))) float v2f;
typedef __attribute__((ext_vector_type(8))) float v8f;

// ---------------------------------------------------------------------------
// Kernel 0: per-point squared norms + zero the edge histogram.
// ---------------------------------------------------------------------------
__global__ void ph_init(const float* __restrict__ pts,
                        float* __restrict__ sq,
                        int* __restrict__ hist) {
    int g = blockIdx.x * blockDim.x + threadIdx.x;
    if (g < B_ * N_) {
        float x = pts[3 * g + 0];
        float y = pts[3 * g + 1];
        float z = pts[3 * g + 2];
        sq[g] = x * x + y * y + z * z;
    }
    if (g < B_ * NBINS) hist[g] = 0;
}

// ---------------------------------------------------------------------------
// Kernel 1: distance tiles via V_WMMA_F32_16X16X4_F32 (K=4, z padded with 0),
// quantize dist -> u8 bucket = ceil(dist*20) with raw v_sqrt_f32, store
// (symmetric matrix: column-major WMMA fragment stored as row-major), and
// histogram buckets for edge counting.
// One wave per 16x16 tile; 8 waves (256 threads) per block.
// Grid: (N/16/8, N/16, B)
// ---------------------------------------------------------------------------
__global__ void ph_dist(const float* __restrict__ pts,
                        const float* __restrict__ sq,
                        unsigned char* __restrict__ bucket,
                        int* __restrict__ hist) {
    __shared__ int lhist[NBINS];
    int tid = threadIdx.x;
    if (tid < NBINS) lhist[tid] = 0;
    __syncthreads();

    int b    = blockIdx.z;
    int i0   = blockIdx.y * 16;
    int wave = tid >> 5;
    int lane = tid & 31;
    int j0   = (blockIdx.x * 8 + wave) * 16;
    int hi   = lane >> 4;   // 0: lanes 0-15 carry K=0,1 ; 1: lanes 16-31 carry K=2,3
    int lm   = lane & 15;

    const float* pbat = pts + (size_t)b * N_ * 3;
    const float* arow = pbat + (size_t)(i0 + lm) * 3;
    const float* brow = pbat + (size_t)(j0 + lm) * 3;

    // A is 16x4 (rows i0..i0+15, K = x,y,z,0) ; B is 4x16 (cols j0..j0+15).
    v2f a, bm;
    a.x  = hi ? arow[2] : arow[0];
    a.y  = hi ? 0.0f    : arow[1];
    bm.x = hi ? brow[2] : brow[0];
    bm.y = hi ? 0.0f    : brow[1];

    v8f acc = {};
    acc = __builtin_amdgcn_wmma_f32_16x16x4_f32(
        /*neg_a=*/false, a, /*neg_b=*/false, bm,
        /*c_mod=*/(short)0, acc, /*reuse_a=*/false, /*reuse_b=*/false);

    // d2[m][n] = |p_i|^2 + |p_j|^2 - 2 * dot ; lane holds col n=lm, rows m=hi*8+v.
    const float* sqb = sq + (size_t)b * N_;
    float sqj = sqb[j0 + lm];
    const float4* sqi4 = (const float4*)(sqb + i0 + hi * 8);
    float4 s0 = sqi4[0], s1 = sqi4[1];
    float sqi[8] = {s0.x, s0.y, s0.z, s0.w, s1.x, s1.y, s1.z, s1.w};

    unsigned int pk0 = 0u, pk1 = 0u;
#pragma unroll
    for (int v = 0; v < 8; ++v) {
        float d2   = sqi[v] + sqj - 2.0f * acc[v];
        // Raw HW sqrt: value is quantized to 1/20-wide buckets, ~1 ulp is plenty.
        float dist = __builtin_amdgcn_sqrtf(fmaxf(d2, 0.0f));
        // adjacency at scale index s (scale = s/20):  dist <= s/20  <=>  bucket <= s
        int bv = (int)ceilf(dist * (float)NSTEPS);
        bv = bv < 0 ? 0 : bv;
        int bin = bv > (NBINS - 1) ? (NBINS - 1) : bv;
        atomicAdd(&lhist[bin], 1);
        bv = bv > 255 ? 255 : bv;
        if (v < 4) pk0 |= ((unsigned int)bv) << ((v & 3) * 8);
        else       pk1 |= ((unsigned int)bv) << ((v & 3) * 8);
    }

    // Symmetric matrix: lane's 8 contiguous rows of column (j0+lm) stored as a
    // contiguous 8-byte row-major run of row (j0+lm). 8B aligned.
    size_t off = (size_t)b * N_ * N_ + (size_t)(j0 + lm) * N_ + (size_t)(i0 + hi * 8);
    *(uint2*)(bucket + off) = make_uint2(pk0, pk1);

    __syncthreads();
    if (tid < NBINS) atomicAdd(&hist[b * NBINS + tid], lhist[tid]);
}

// ---------------------------------------------------------------------------
// Kernel 2: connected components per (scale, batch). 1024 threads, labels in
// LDS. Adjacency row (1024 bits) built ONCE into 32 VGPRs. Per iteration:
//   - wave w (covering labels w*32..w*32+31) computes its block-min via 5
//     __shfl_xor steps and publishes it to LDS,
//   - all-ones mask words take the block-min with ONE LDS read (dense case),
//     zero words are free, partial words pay the bit loop,
//   - min-propagate + pointer doubling, early exit at fixpoint (identical
//     result to the reference's fixed 12 iterations, which are identity after
//     convergence).
// Grid: (NSTEPS, B_), block 1024.
// ---------------------------------------------------------------------------
__global__ void __launch_bounds__(1024) ph_cc(const unsigned char* __restrict__ bucket,
                                              int* __restrict__ comp) {
    __shared__ int lbl[N_];
    __shared__ int bmin[N_ / 32];
    __shared__ int cnt;
    __shared__ int chg;
    int s = blockIdx.x;
    int b = blockIdx.y;
    int i = threadIdx.x;
    lbl[i] = i;
    if (i == 0) cnt = 0;

    const uint4* row = (const uint4*)(bucket + (size_t)b * N_ * N_ + (size_t)i * N_);
    unsigned int su = (unsigned int)s;
    __builtin_prefetch(row, 0, 0);

    // One pass: u8 buckets -> 1024-bit adjacency mask in registers.
    unsigned int mask[32];
#pragma unroll
    for (int w = 0; w < 32; ++w) {
        unsigned int bits = 0u;
#pragma unroll
        for (int h = 0; h < 2; ++h) {
            uint4 v = row[w * 2 + h];
            unsigned int wd[4] = {v.x, v.y, v.z, v.w};
#pragma unroll
            for (int q = 0; q < 4; ++q) {
#pragma unroll
                for (int k = 0; k < 4; ++k) {
                    if (((wd[q] >> (k * 8)) & 0xffu) <= su)
                        bits |= 1u << (h * 16 + q * 4 + k);
                }
            }
        }
        mask[w] = bits;
    }
    __syncthreads();

    for (int it = 0; it < 12; ++it) {
        if (i == 0) chg = 0;
        int old = lbl[i];

        // Block-min of this wave's 32 labels (wave w owns labels w*32..+31).
        int t = old;
#pragma unroll
        for (int o = 16; o > 0; o >>= 1) {
            int u = __shfl_xor(t, o, 32);
            t = t < u ? t : u;
        }
        if ((i & 31) == 0) bmin[i >> 5] = t;
        __syncthreads();                 // bmin ready, chg reset visible

        int m = old;
#pragma unroll
        for (int w = 0; w < 32; ++w) {
            unsigned int bits = mask[w];
            if (bits == 0xffffffffu) {   // fully connected to this block
                int l = bmin[w];
                m = m < l ? m : l;
            } else if (bits) {
#pragma unroll
                for (int k = 0; k < 32; ++k) {
                    if (bits & (1u << k)) {
                        int l = lbl[w * 32 + k];
                        m = m < l ? m : l;
                    }
                }
            }
        }
        __syncthreads();
        lbl[i] = m;          // lbl = min(lbl, neighbor-min)  (diag always adjacent)
        __syncthreads();
        int pd = lbl[m];     // pointer doubling: lbl = lbl[lbl]
        __syncthreads();
        lbl[i] = pd;
        if (pd != old) atomicOr(&chg, 1);
        __syncthreads();
        if (chg == 0) break; // fixpoint: remaining reference iterations are identity
        __syncthreads();     // keep flag reset (next iter) after everyone's read
    }

    if (lbl[i] == i) atomicAdd(&cnt, 1);
    __syncthreads();
    if (i == 0) comp[b * NSTEPS + s] = cnt;
}

// ---------------------------------------------------------------------------
// Kernel 3: Betti numbers + persistence diagram.
// diag (2,B,20,2) floats at out[0..1279]; betti (2,B,20) int32 at out+1280.
// ---------------------------------------------------------------------------
__global__ void ph_final(const int* __restrict__ hist,
                         const int* __restrict__ comp,
                         float* __restrict__ out) {
    int t = threadIdx.x;
    if (t >= B_ * NSTEPS) return;
    int b = t / NSTEPS, s = t % NSTEPS;

    int cum = 0;
    for (int v = 0; v <= s; ++v) cum += hist[b * NBINS + v];
    int edges = (cum - N_) / 2;                 // (sum(adj) - n) / 2
    int nc = comp[b * NSTEPS + s];              // Betti-0
    int b1 = edges - N_ + nc;                   // cycle rank
    b1 = b1 > 0 ? b1 : 0;

    float birth = (float)s * (1.0f / (float)NSTEPS);

    int* betti = (int*)(out + 2 * B_ * NSTEPS * 2);   // after 1280 floats
    betti[0 * B_ * NSTEPS + b * NSTEPS + s] = nc;
    betti[1 * B_ * NSTEPS + b * NSTEPS + s] = b1;

    int i0 = ((0 * B_ + b) * NSTEPS + s) * 2;
    int i1 = ((1 * B_ + b) * NSTEPS + s) * 2;
    out[i0]     = nc > 0 ? birth : 0.0f;
    out[i0 + 1] = nc > 0 ? 1.0f : 0.0f;
    out[i1]     = b1 > 0 ? birth : 0.0f;
    out[i1 + 1] = b1 > 0 ? 1.0f : 0.0f;
}

// ---------------------------------------------------------------------------
extern "C" void kernel_launch(void* const* d_in, const int* in_sizes, int n_in,
                              void* d_out, int out_size, void* d_ws, size_t ws_size,
                              hipStream_t stream) {
    const float* pts = (const float*)d_in[0];        // (16,1024,3) f32
    float* out = (float*)d_out;

    char* ws = (char*)d_ws;
    unsigned char* bucket = (unsigned char*)ws;                       // 16 MB
    float* sq   = (float*)(ws + (size_t)B_ * N_ * N_);                // 64 KB
    int*   hist = (int*)(ws + (size_t)B_ * N_ * N_ + (size_t)B_ * N_ * sizeof(float));
    int*   comp = hist + B_ * NBINS;

    ph_init<<<(B_ * N_ + 255) / 256, 256, 0, stream>>>(pts, sq, hist);
    ph_dist<<<dim3(N_ / 16 / 8, N_ / 16, B_), 256, 0, stream>>>(pts, sq, bucket, hist);
    ph_cc<<<dim3(NSTEPS, B_), 1024, 0, stream>>>(bucket, comp);
    ph_final<<<1, 320, 0, stream>>>(hist, comp, out);
}